// SDPAttention_24592982736977
// MI455X (gfx1250) — compile-verified
//
#include <hip/hip_runtime.h>
#include <cstddef>
#include <cstdint>

typedef float v2f __attribute__((ext_vector_type(2)));
typedef float v4f __attribute__((ext_vector_type(4)));
typedef float v8f __attribute__((ext_vector_type(8)));

#define KD 128
#define NWAVES 8

// ---------- DPP16 16-lane sum reduction (all-VALU, no LDS) ----------
// Reduction tree over each 16-lane row; every step pairs disjoint halves of
// the not-yet-merged groups, so the final value is the row sum replicated to
// all 16 lanes.  quad_perm[1,0,3,2]=0xB1, quad_perm[2,3,0,1]=0x4E,
// row_half_mirror=0x141, row_mirror=0x140.
template <int CTRL>
__device__ __forceinline__ float dpp_add_f32(float x) {
  int y = __builtin_amdgcn_mov_dpp(__float_as_int(x), CTRL, 0xF, 0xF, true);
  return x + __int_as_float(y);
}

__device__ __forceinline__ float row16_sum(float x) {
  x = dpp_add_f32<0xB1>(x);    // xor 1 within quads
  x = dpp_add_f32<0x4E>(x);    // xor 2 within quads
  x = dpp_add_f32<0x141>(x);   // pairs quads within 8
  x = dpp_add_f32<0x140>(x);   // pairs octets within 16
  return x;
}

// Cross-half (lane ^ 16) exchange. Prefer VALU v_permlanex16 over ds_bpermute.
__device__ __forceinline__ float xhalf(float x) {
#if __has_builtin(__builtin_amdgcn_permlanex16)
  int y = __builtin_amdgcn_permlanex16(__float_as_int(x), __float_as_int(x),
                                       0x76543210, 0xFEDCBA98,
                                       /*fi=*/false, /*bound_ctrl=*/false);
  return __int_as_float(y);
#else
  return __shfl_xor(x, 16, 32);
#endif
}

__device__ __forceinline__ int lower_bound_i32(const int* __restrict__ a, int n, int key) {
  int lo = 0, hi = n;
  while (lo < hi) {
    int mid = (lo + hi) >> 1;
    if (a[mid] < key) lo = mid + 1; else hi = mid;
  }
  return lo;
}

// Zero-fill the whole [V,128] output with non-temporal 16B stores (~512MB).
__global__ void __launch_bounds__(256) zero_fill_kernel(v4f* __restrict__ out, long long n4) {
  long long i = (long long)blockIdx.x * blockDim.x + threadIdx.x;
  const long long stride = (long long)gridDim.x * blockDim.x;
  v4f z = {0.0f, 0.0f, 0.0f, 0.0f};
  for (; i < n4; i += stride) __builtin_nontemporal_store(z, out + i);
}

// One workgroup (8 wave32s) per segment b. Single streaming pass over the
// segment's rows with an online (flash-style) softmax; weighted pooling is
// accumulated with V_WMMA_F32_16X16X4_F32.
__global__ void __launch_bounds__(256) seg_attn_kernel(const float* __restrict__ X,
                                                       const float* __restrict__ Q,
                                                       const int* __restrict__ bidx,
                                                       float* __restrict__ out,
                                                       int V) {
  const int b    = blockIdx.x;
  const int tid  = threadIdx.x;
  const int lane = tid & 31;
  const int wav  = tid >> 5;
  const int hh   = lane >> 4;   // which 16-lane half
  const int fl   = lane & 15;   // feature index within a 16-wide group

  __shared__ float lds_pool[NWAVES * KD];
  __shared__ float lds_m[NWAVES];
  __shared__ float lds_s[NWAVES];

  // Contiguous segment bounds via binary search on the sorted batch_index.
  const int seg_start = lower_bound_i32(bidx, V, b);
  const int seg_end   = lower_bound_i32(bidx, V, b + 1);
  const int seg_len   = seg_end - seg_start;

  // Pre-scale Q[b] by 1/sqrt(128); q[g] = Q[b, g*16 + fl] (same in both halves).
  const float inv_sqrt_d = 0.08838834764831845f;
  float q[8];
  #pragma unroll
  for (int g = 0; g < 8; ++g) q[g] = Q[b * KD + g * 16 + fl] * inv_sqrt_d;

  float m_run = -__builtin_inff();
  float s_run = 0.0f;
  const v8f vzero = {0.f, 0.f, 0.f, 0.f, 0.f, 0.f, 0.f, 0.f};
  v8f acc[8];                       // 8 groups x 16 features, pooled sums
  #pragma unroll
  for (int g = 0; g < 8; ++g) acc[g] = vzero;

  const int ntiles = (seg_len + 15) >> 4;
  for (int t = wav; t < ntiles; t += NWAVES) {
    const int base = seg_start + t * 16;

    // ---- Load A operands directly in WMMA shape: A[c][g] is the even-aligned
    // VGPR pair for V_WMMA A(16x4):  .x = K0|K2 (nodes +0/+2),
    // .y = K1|K3 (nodes +1/+3), lanes = feature m within group g.
    v2f A[4][8];
    #pragma unroll
    for (int c = 0; c < 4; ++c) {
      int node0 = base + c * 4 + hh * 2;       // this half's even node
      int nc0 = node0     < seg_end ? node0     : (seg_end - 1);
      int nc1 = node0 + 1 < seg_end ? node0 + 1 : (seg_end - 1);
      const float* rp0 = X + (size_t)nc0 * KD + fl;
      const float* rp1 = X + (size_t)nc1 * KD + fl;
      #pragma unroll
      for (int g = 0; g < 8; ++g) {
        A[c][g].x = rp0[g * 16];
        A[c][g].y = rp1[g * 16];
      }
    }

    // ---- Scores: per-lane partial dot + DPP16 row reduction (no LDS).
    // p[c][0]: lanes0-15 = score(n0), lanes16-31 = score(n2); p[c][1]: n1|n3.
    float p[4][2];
    #pragma unroll
    for (int c = 0; c < 4; ++c) {
      float s0 = 0.0f, s1 = 0.0f;
      #pragma unroll
      for (int g = 0; g < 8; ++g) {
        s0 = fmaf(A[c][g].x, q[g], s0);
        s1 = fmaf(A[c][g].y, q[g], s1);
      }
      s0 = row16_sum(s0);
      s1 = row16_sum(s1);
      int node0 = base + c * 4 + hh * 2;
      p[c][0] = (node0     < seg_end) ? s0 : -__builtin_inff();
      p[c][1] = (node0 + 1 < seg_end) ? s1 : -__builtin_inff();
    }

    // ---- Tile max (across both halves).
    float tmax = p[0][0];
    #pragma unroll
    for (int c = 0; c < 4; ++c) {
      tmax = fmaxf(tmax, p[c][0]);
      tmax = fmaxf(tmax, p[c][1]);
    }
    tmax = fmaxf(tmax, xhalf(tmax));

    const float m_new = fmaxf(m_run, tmax);
    // Rescale only when the running max actually moved (rare after warm-up).
    // tmax/m_run are wave-uniform -> force a scalar branch; EXEC stays all-1s.
    if (__builtin_amdgcn_readfirstlane((int)(tmax > m_run))) {
      const float sc = __expf(m_run - m_new);   // exp(-inf)=0 on first tile
      s_run *= sc;
      #pragma unroll
      for (int g = 0; g < 8; ++g) {
        #pragma unroll
        for (int r = 0; r < 8; ++r) acc[g][r] *= sc;
      }
    }

    // ---- Weights. Half-replicated exp registers ARE the ISA B(4x16) layout:
    // Bop[c].x = w0|w2, Bop[c].y = w1|w3.
    v2f Bop[4];
    float wsum = 0.0f;
    #pragma unroll
    for (int c = 0; c < 4; ++c) {
      float w0 = __expf(p[c][0] - m_new);
      float w1 = __expf(p[c][1] - m_new);
      wsum += w0 + w1;
      Bop[c].x = w0;
      Bop[c].y = w1;
    }
    wsum += xhalf(wsum);
    s_run += wsum;
    m_run = m_new;

    // ---- Pooling: acc[g] += A(c,g) x B(c)  (V_WMMA_F32_16X16X4_F32).
    // C[m,n] = sum_k X[node_k, g*16+m] * w_k for every n.
    #pragma unroll
    for (int c = 0; c < 4; ++c) {
      #pragma unroll
      for (int g = 0; g < 8; ++g) {
        acc[g] = __builtin_amdgcn_wmma_f32_16x16x4_f32(
            /*neg_a=*/false, A[c][g], /*neg_b=*/false, Bop[c],
            /*c_mod=*/(short)0, acc[g], /*reuse_a=*/false, /*reuse_b=*/false);
      }
    }
  }

  // ---- Export per-wave state: column N=0 of C lives in lane 0 (M=r) and
  // lane 16 (M=8+r) of each accumulator VGPR.
  if (lane == 0 || lane == 16) {
    const int hi = (lane == 16) ? 8 : 0;
    #pragma unroll
    for (int g = 0; g < 8; ++g) {
      #pragma unroll
      for (int r = 0; r < 8; ++r)
        lds_pool[wav * KD + g * 16 + hi + r] = acc[g][r];
    }
  }
  if (lane == 0) {
    lds_m[wav] = m_run;
    lds_s[wav] = s_run;
  }
  __syncthreads();

  // ---- Cross-wave merge (flash combine) + final normalize; 128 threads.
  if (tid < KD) {
    float M = -__builtin_inff();
    #pragma unroll
    for (int w = 0; w < NWAVES; ++w) M = fmaxf(M, lds_m[w]);
    float S = 0.0f, val = 0.0f;
    if (M > -__builtin_inff()) {
      #pragma unroll
      for (int w = 0; w < NWAVES; ++w) {
        float e = __expf(lds_m[w] - M);
        S   = fmaf(e, lds_s[w], S);
        val = fmaf(e, lds_pool[w * KD + tid], val);
      }
    }
    out[(size_t)b * KD + tid] = (S > 0.0f) ? (val / S) : 0.0f;
  }
}

extern "C" void kernel_launch(void* const* d_in, const int* in_sizes, int n_in,
                              void* d_out, int out_size, void* d_ws, size_t ws_size,
                              hipStream_t stream) {
  const float* X    = (const float*)d_in[0];   // node_feats [V,128] f32
  const float* Q    = (const float*)d_in[1];   // Q [B,128] f32
  const int*   bidx = (const int*)d_in[2];     // batch_index [V] i32 (sorted)
  float* out = (float*)d_out;                  // [V,128] f32

  const int V = in_sizes[0] / KD;
  const int B = in_sizes[1] / KD;

  const long long n4 = (long long)out_size / 4;
  zero_fill_kernel<<<4096, 256, 0, stream>>>((v4f*)out, n4);
  seg_attn_kernel<<<B, 256, 0, stream>>>(X, Q, bidx, out, V);
}